// GCNWithMolFeatures_86019605004840
// MI455X (gfx1250) — compile-verified
//
#include <hip/hip_runtime.h>

typedef __attribute__((ext_vector_type(2))) float v2f;
typedef __attribute__((ext_vector_type(8))) float v8f;

__global__ void k_zero(float* __restrict__ p, int n) {
  int i = blockIdx.x * blockDim.x + threadIdx.x;
  if (i < n) p[i] = 0.0f;
}

__global__ void k_deg(const int* __restrict__ dst, int* __restrict__ deg, int E) {
  int i = blockIdx.x * blockDim.x + threadIdx.x;
  if (i < E) atomicAdd(&deg[dst[i]], 1);
}

__global__ void k_dinv(const int* __restrict__ deg, float* __restrict__ dinv, int N) {
  int i = blockIdx.x * blockDim.x + threadIdx.x;
  if (i < N) dinv[i] = rsqrtf((float)deg[i] + 1.0f);
}

__global__ void k_enorm(const int* __restrict__ src, const int* __restrict__ dst,
                        const float* __restrict__ dinv, float* __restrict__ enorm, int E) {
  int i = blockIdx.x * blockDim.x + threadIdx.x;
  if (i < E) enorm[i] = dinv[src[i]] * dinv[dst[i]];
}

// Fused: hW = (RELU ? relu(X) : X) @ W[64x64]   (fp32 matrix core path)
//        agg = bias + hW * dinv^2               (GCN self-loop term, from registers)
// Block = 128 threads = 4 waves; block owns a 16-row tile, wave w owns cols
// [16w, 16w+16). 16 x v_wmma_f32_16x16x4_f32 with A/B fragments preloaded.
// Partial tiles: loads clamp the row (EXEC stays all-ones for WMMA), stores guard.
template <bool RELU>
__global__ void k_gemm64(const float* __restrict__ X, const float* __restrict__ W,
                         const float* __restrict__ dinv, const float* __restrict__ bias,
                         float* __restrict__ hW, float* __restrict__ agg, int nrows) {
  int wave = threadIdx.x >> 5;
  int lane = threadIdx.x & 31;
  int m0 = blockIdx.x * 16;
  int c0 = wave * 16;
  int half = lane >> 4;   // 0: K pair {0,1}; 1: K pair {2,3}
  int l15 = lane & 15;

  int arow = m0 + l15;
  if (arow >= nrows) arow = nrows - 1;           // clamp: store-side guard is enough
  const v2f* ap = (const v2f*)(X + (size_t)arow * 64 + half * 2);  // 8B aligned
  const float* bp = W + (size_t)(half * 2) * 64 + c0 + l15;

  v2f afrag[16];
  v2f bfrag[16];
#pragma unroll
  for (int i = 0; i < 16; ++i) {
    v2f a = ap[i * 2];                            // A[row, 4i + 2*half + {0,1}]
    if (RELU) { a.x = fmaxf(a.x, 0.0f); a.y = fmaxf(a.y, 0.0f); }
    afrag[i] = a;
    v2f b;                                        // B[4i + 2*half + {0,1}, c0+l15]
    b.x = bp[(size_t)(i * 4) * 64];
    b.y = bp[(size_t)(i * 4 + 1) * 64];
    bfrag[i] = b;
  }

  v8f acc = {};
#pragma unroll
  for (int i = 0; i < 16; ++i)
    acc = __builtin_amdgcn_wmma_f32_16x16x4_f32(false, afrag[i], false, bfrag[i],
                                                (short)0, acc, false, false);

  // C/D layout: VGPR v, lanes 0-15 -> row v, lanes 16-31 -> row v+8; col = lane&15
  int rbase = m0 + half * 8;
  float bv = bias[c0 + l15];
#pragma unroll
  for (int v = 0; v < 8; ++v) {
    int r = rbase + v;
    if (r < nrows) {
      float hv = acc[v];
      float di = dinv[r];
      size_t idx = (size_t)r * 64 + c0 + l15;
      hW[idx] = hv;
      agg[idx] = bv + hv * di * di;
    }
  }
}

// agg[dst,f] += hW[src,f] * enorm[e]; thread = (edge, feature). Each wave's 32
// lanes share one edge -> src/dst/enorm coalesce to broadcast requests; the f32
// atomics resolve in the 192MB L2 (whole working set is L2-resident).
__global__ void k_edge(const float* __restrict__ hW, const int* __restrict__ src,
                       const int* __restrict__ dst, const float* __restrict__ enorm,
                       float* __restrict__ agg, int E) {
  int i = blockIdx.x * blockDim.x + threadIdx.x;
  int e = i >> 6, f = i & 63;
  if (e >= E) return;
  float v = hW[(size_t)src[e] * 64 + f] * enorm[e];
  atomicAdd(&agg[(size_t)dst[e] * 64 + f], v);
}

// segment sum/max/count over batch; v >= 0 post-ReLU so uint-bit atomicMax is
// order-preserving and the 0-init matches the reference's empty-graph masking.
__global__ void k_pool(const float* __restrict__ hpre, const int* __restrict__ batch,
                       float* __restrict__ psum, unsigned* __restrict__ pmax,
                       int* __restrict__ cnt, int total) {
  int i = blockIdx.x * blockDim.x + threadIdx.x;
  if (i >= total) return;
  int n = i >> 6, f = i & 63;
  float v = fmaxf(hpre[i], 0.0f);
  int g = batch[n];
  atomicAdd(&psum[(size_t)g * 64 + f], v);
  atomicMax(&pmax[(size_t)g * 64 + f], __float_as_uint(v));
  if (f == 0) atomicAdd(&cnt[g], 1);
}

// out[g] = relu([max | mean | mol] @ fcW1 + fcb1) @ fcW2 + fcb2
__global__ void k_fc(const float* __restrict__ psum, const unsigned* __restrict__ pmax,
                     const int* __restrict__ cnt, const float* __restrict__ mol,
                     const float* __restrict__ fcW1, const float* __restrict__ fcb1,
                     const float* __restrict__ fcW2, const float* __restrict__ fcb2,
                     float* __restrict__ out) {
  __shared__ float comb[256];
  __shared__ float red[64];
  int g = blockIdx.x, t = threadIdx.x;  // 64 threads
  float invc = 1.0f / fmaxf((float)cnt[g], 1.0f);
  comb[t]       = __uint_as_float(pmax[(size_t)g * 64 + t]);
  comb[64 + t]  = psum[(size_t)g * 64 + t] * invc;
  comb[128 + t] = mol[(size_t)g * 128 + t];
  comb[192 + t] = mol[(size_t)g * 128 + 64 + t];
  __syncthreads();
  float acc = fcb1[t];
#pragma unroll 8
  for (int i = 0; i < 256; ++i) acc = fmaf(comb[i], fcW1[(size_t)i * 64 + t], acc);
  red[t] = fmaxf(acc, 0.0f) * fcW2[t];
  __syncthreads();
  for (int s = 32; s > 0; s >>= 1) {
    if (t < s) red[t] += red[t + s];
    __syncthreads();
  }
  if (t == 0) out[g] = red[0] + fcb2[0];
}

extern "C" void kernel_launch(void* const* d_in, const int* in_sizes, int n_in,
                              void* d_out, int out_size, void* d_ws, size_t ws_size,
                              hipStream_t stream) {
  const float* x    = (const float*)d_in[0];
  const int*   eidx = (const int*)d_in[1];
  const int*   batch= (const int*)d_in[2];
  const float* mol  = (const float*)d_in[3];
  const float* W1   = (const float*)d_in[4];  const float* b1 = (const float*)d_in[5];
  const float* W2   = (const float*)d_in[6];  const float* b2 = (const float*)d_in[7];
  const float* W3   = (const float*)d_in[8];  const float* b3 = (const float*)d_in[9];
  const float* fcW1 = (const float*)d_in[10]; const float* fcb1 = (const float*)d_in[11];
  const float* fcW2 = (const float*)d_in[12]; const float* fcb2 = (const float*)d_in[13];
  float* out = (float*)d_out;

  const int N = in_sizes[0] / 64;
  const int E = in_sizes[1] / 2;
  const int G = in_sizes[3] / 128;
  const int* src = eidx;
  const int* dst = eidx + E;

  // workspace carve-out (256B aligned)
  char* ws = (char*)d_ws;
  size_t off = 0;
  auto carve = [&](size_t bytes) -> void* {
    void* p = ws + off;
    off += (bytes + 255) & ~(size_t)255;
    return p;
  };
  int*      deg   = (int*)     carve((size_t)N * 4);
  float*    dinv  = (float*)   carve((size_t)N * 4);
  float*    enorm = (float*)   carve((size_t)E * 4);
  float*    hW    = (float*)   carve((size_t)N * 64 * 4);
  float*    bufA  = (float*)   carve((size_t)N * 64 * 4);
  float*    bufB  = (float*)   carve((size_t)N * 64 * 4);
  float*    psum  = (float*)   carve((size_t)G * 64 * 4);
  unsigned* pmax  = (unsigned*)carve((size_t)G * 64 * 4);
  int*      cnt   = (int*)     carve((size_t)G * 4);
  (void)ws_size; (void)n_in; (void)out_size;

  const int TB = 256;
  auto cdiv = [](long long a, long long b) { return (int)((a + b - 1) / b); };
  const int NE = N * 64;                   // 6.4M
  const long long EE = (long long)E * 64;  // 64M
  const int gtiles = cdiv(N, 16);

  // --- init / normalization ---
  k_zero<<<cdiv(N, TB), TB, 0, stream>>>((float*)deg, N);
  k_zero<<<cdiv(G * 64, TB), TB, 0, stream>>>(psum, G * 64);
  k_zero<<<cdiv(G * 64, TB), TB, 0, stream>>>((float*)pmax, G * 64);
  k_zero<<<cdiv(G, TB), TB, 0, stream>>>((float*)cnt, G);

  k_deg  <<<cdiv(E, TB), TB, 0, stream>>>(dst, deg, E);
  k_dinv <<<cdiv(N, TB), TB, 0, stream>>>(deg, dinv, N);
  k_enorm<<<cdiv(E, TB), TB, 0, stream>>>(src, dst, dinv, enorm, E);

  // --- layer 1: x -> bufA ---
  k_gemm64<false><<<gtiles, 128, 0, stream>>>(x, W1, dinv, b1, hW, bufA, N);
  k_edge<<<cdiv(EE, TB), TB, 0, stream>>>(hW, src, dst, enorm, bufA, E);

  // --- layer 2: relu(bufA) -> bufB ---
  k_gemm64<true><<<gtiles, 128, 0, stream>>>(bufA, W2, dinv, b2, hW, bufB, N);
  k_edge<<<cdiv(EE, TB), TB, 0, stream>>>(hW, src, dst, enorm, bufB, E);

  // --- layer 3: relu(bufB) -> bufA ---
  k_gemm64<true><<<gtiles, 128, 0, stream>>>(bufB, W3, dinv, b3, hW, bufA, N);
  k_edge<<<cdiv(EE, TB), TB, 0, stream>>>(hW, src, dst, enorm, bufA, E);

  // --- readout + MLP ---
  k_pool<<<cdiv(NE, TB), TB, 0, stream>>>(bufA, batch, psum, pmax, cnt, NE);
  k_fc  <<<G, 64, 0, stream>>>(psum, pmax, cnt, mol, fcW1, fcb1, fcW2, fcb2, out);
}